// GCNLayer_4526895530478
// MI455X (gfx1250) — compile-verified
//
#include <hip/hip_runtime.h>

#define IN_SIZE 128
#define OUT_SIZE 128
#define K2STRIDE 288  // floats per k-pair row; 288 % 64 == 32 -> half-waves hit disjoint banks

typedef __attribute__((ext_vector_type(2))) float v2f;
typedef __attribute__((ext_vector_type(8))) float v8f;

__global__ __launch_bounds__(256) void zero_f32(float* __restrict__ p, size_t n) {
  size_t i = (size_t)blockIdx.x * blockDim.x + threadIdx.x;
  size_t stride = (size_t)gridDim.x * blockDim.x;
  for (; i < n; i += stride) p[i] = 0.0f;
}

// One wave32 per edge: agg[dst] += h[src] * norm[src].  Lane covers 4 channels.
__global__ __launch_bounds__(256) void spmm_edges(
    const float* __restrict__ h, const float* __restrict__ norm,
    const int* __restrict__ src, const int* __restrict__ dst,
    float* __restrict__ agg, int n_edges)
{
  int e = (int)(((size_t)blockIdx.x * blockDim.x + threadIdx.x) >> 5);
  if (e >= n_edges) return;
  int lane = threadIdx.x & 31;
  int s = src[e];           // wave-uniform -> scalarized
  int d = dst[e];
  float ns = norm[s];
  float4 v = ((const float4*)(h + (size_t)s * IN_SIZE))[lane];
  float* ap = agg + (size_t)d * IN_SIZE + lane * 4;
  atomicAdd(ap + 0, v.x * ns);
  atomicAdd(ap + 1, v.y * ns);
  atomicAdd(ap + 2, v.z * ns);
  atomicAdd(ap + 3, v.w * ns);
}

// y = relu((agg*norm) @ W + b), in-place over agg rows; per-channel sum/sumsq via
// LDS reduction then one global atomic per channel per block.
// Block = 128 threads = 4 waves; each wave owns a 16-row M tile (64 rows / block).
// WMMA f32 16x16x4: 8 N-tiles x 32 K-steps per wave.
// W is staged in LDS K-pair interleaved: WB[k2][n] = {W[2k2][n], W[2k2+1][n]} so a
// lane's B operand is a single conflict-free ds_load_b64 straight into a VGPR pair.
__global__ __launch_bounds__(128) void gemm_relu_stats(
    const float* agg, const float* __restrict__ norm,
    const float* __restrict__ W, const float* __restrict__ bias,
    float* y, float* __restrict__ g_sum, float* __restrict__ g_sumsq,
    int n_nodes)
{
  __shared__ float WB[64 * K2STRIDE];   // 72 KB
  __shared__ float s_sum[128];
  __shared__ float s_sq[128];
  int tid = threadIdx.x;

  // cooperative pair-interleaved load of W[128][128] into LDS
  for (int i = tid; i < 8192; i += 128) {
    int k2 = i >> 7;          // K pair index 0..63
    int n  = i & 127;         // column
    float2 w2;
    w2.x = W[(2 * k2 + 0) * OUT_SIZE + n];
    w2.y = W[(2 * k2 + 1) * OUT_SIZE + n];
    *(float2*)(WB + k2 * K2STRIDE + 2 * n) = w2;
  }
  s_sum[tid] = 0.0f;
  s_sq[tid]  = 0.0f;
  __syncthreads();

  int wave   = tid >> 5;
  int lane   = tid & 31;
  int laneLo = lane & 15;
  int laneHi = lane >> 4;
  int mbase  = blockIdx.x * 64 + wave * 16;

  if (mbase < n_nodes) {                 // wave-uniform: EXEC stays all-ones inside
    // A layout (16x4 f32): lanes 0-15 -> M=lane, K={0,1}; lanes 16-31 -> M=lane-16, K={2,3}
    int arow = mbase + laneLo;
    if (arow > n_nodes - 1) arow = n_nodes - 1;
    float nr = norm[arow];
    const float* ap = agg + (size_t)arow * IN_SIZE + 2 * laneHi;

    v8f acc[8] = {};                     // 8 N-tiles of 16 columns each

    for (int k = 0; k < 32; ++k) {
      v2f a = *(const v2f*)(ap + 4 * k);          // global b64 load
      a.x *= nr;
      a.y *= nr;
      // B (4x16 KxN) mirrors A's K striping: this lane needs K pair k2 = 2k + laneHi
      const float* wb = WB + (size_t)(2 * k + laneHi) * K2STRIDE + 2 * laneLo;
      #pragma unroll
      for (int n = 0; n < 8; ++n) {
        v2f b = *(const v2f*)(wb + n * 32);       // single ds_load_b64, conflict-free
        acc[n] = __builtin_amdgcn_wmma_f32_16x16x4_f32(
            false, a, false, b, (short)0, acc[n], false, false);
      }
    }

    // Epilogue: D VGPR v holds row mbase+v (lanes 0-15) / mbase+v+8 (lanes 16-31),
    // column n*16+laneLo.  Bias + ReLU, store in place, reduce BN stats.
    #pragma unroll
    for (int n = 0; n < 8; ++n) {
      int col = n * 16 + laneLo;
      float bc = bias[col];
      float ps = 0.0f, pq = 0.0f;
      #pragma unroll
      for (int v = 0; v < 8; ++v) {
        int row = mbase + v + 8 * laneHi;
        float val = acc[n][v] + bc;
        val = val > 0.0f ? val : 0.0f;
        if (row < n_nodes) {
          y[(size_t)row * OUT_SIZE + col] = val;
          ps += val;
          pq += val * val;
        }
      }
      atomicAdd(&s_sum[col], ps);        // ds_add_f32
      atomicAdd(&s_sq[col],  pq);
    }
  }
  __syncthreads();
  atomicAdd(&g_sum[tid],   s_sum[tid]);  // one global atomic per channel per block
  atomicAdd(&g_sumsq[tid], s_sq[tid]);
}

__global__ __launch_bounds__(128) void bn_finalize(
    const float* __restrict__ g_sum, const float* __restrict__ g_sumsq,
    const float* __restrict__ gamma, const float* __restrict__ beta,
    float* __restrict__ scale, float* __restrict__ shift, float inv_n)
{
  int c = threadIdx.x;
  float mean = g_sum[c] * inv_n;
  float var  = g_sumsq[c] * inv_n - mean * mean;   // biased variance
  float sc   = gamma[c] * rsqrtf(var + 1e-5f);
  scale[c] = sc;
  shift[c] = beta[c] - mean * sc;
}

// In-place BN apply + sum_nodes pooling. graph_ids are sorted, so each wave walks
// 16 consecutive nodes accumulating phis in registers, flushing atomics only on
// graph-ID change (wave-uniform condition).
__global__ __launch_bounds__(256) void bn_apply_pool(
    float* out_x, float* out_phis,
    const float* __restrict__ scale, const float* __restrict__ shift,
    const int* __restrict__ gids, int n_nodes)
{
  const int NPW = 16;
  int w = (int)(((size_t)blockIdx.x * blockDim.x + threadIdx.x) >> 5);
  int lane = threadIdx.x & 31;
  int node0 = w * NPW;
  if (node0 >= n_nodes) return;
  int col = lane * 4;
  float4 sc = *(const float4*)(scale + col);
  float4 sh = *(const float4*)(shift + col);
  float4 acc = make_float4(0.f, 0.f, 0.f, 0.f);
  int cur_g = gids[node0];               // wave-uniform
  int nend = node0 + NPW;
  if (nend > n_nodes) nend = n_nodes;
  for (int nd = node0; nd < nend; ++nd) {
    int g = gids[nd];
    if (g != cur_g) {
      float* pp = out_phis + (size_t)cur_g * OUT_SIZE + col;
      atomicAdd(pp + 0, acc.x); atomicAdd(pp + 1, acc.y);
      atomicAdd(pp + 2, acc.z); atomicAdd(pp + 3, acc.w);
      acc = make_float4(0.f, 0.f, 0.f, 0.f);
      cur_g = g;
    }
    float4* xp = (float4*)(out_x + (size_t)nd * OUT_SIZE + col);
    float4 v = *xp;
    v.x = v.x * sc.x + sh.x;
    v.y = v.y * sc.y + sh.y;
    v.z = v.z * sc.z + sh.z;
    v.w = v.w * sc.w + sh.w;
    *xp = v;
    acc.x += v.x; acc.y += v.y; acc.z += v.z; acc.w += v.w;
  }
  float* pp = out_phis + (size_t)cur_g * OUT_SIZE + col;
  atomicAdd(pp + 0, acc.x); atomicAdd(pp + 1, acc.y);
  atomicAdd(pp + 2, acc.z); atomicAdd(pp + 3, acc.w);
}

extern "C" void kernel_launch(void* const* d_in, const int* in_sizes, int n_in,
                              void* d_out, int out_size, void* d_ws, size_t ws_size,
                              hipStream_t stream)
{
  const float* h     = (const float*)d_in[0];
  const float* norm  = (const float*)d_in[1];
  const float* W     = (const float*)d_in[2];
  const float* bias  = (const float*)d_in[3];
  const float* gamma = (const float*)d_in[4];
  const float* beta  = (const float*)d_in[5];
  const int*   src   = (const int*)d_in[6];
  const int*   dst   = (const int*)d_in[7];
  const int*   gids  = (const int*)d_in[8];

  int n_nodes = in_sizes[1];   // norm is [N,1]
  int n_edges = in_sizes[6];

  float* out_x    = (float*)d_out;
  float* out_phis = out_x + (size_t)n_nodes * OUT_SIZE;

  float* g_sum   = (float*)d_ws;       // [128]
  float* g_sumsq = g_sum + 128;        // [128]
  float* scale   = g_sum + 256;        // [128]
  float* shift   = g_sum + 384;        // [128]

  // 1. zero the full output (x region doubles as the SpMM accumulator) + stats
  zero_f32<<<2048, 256, 0, stream>>>(out_x, (size_t)out_size);
  zero_f32<<<1, 256, 0, stream>>>(g_sum, (size_t)256);

  // 2. edge scatter: agg (= out_x region) += h[src]*norm[src]
  spmm_edges<<<(n_edges + 7) / 8, 256, 0, stream>>>(h, norm, src, dst, out_x, n_edges);

  // 3. in-place GEMM + bias + ReLU + BN-stat reduction (WMMA f32 16x16x4)
  gemm_relu_stats<<<(n_nodes + 63) / 64, 128, 0, stream>>>(
      out_x, norm, W, bias, out_x, g_sum, g_sumsq, n_nodes);

  // 4. fold stats into per-channel scale/shift
  bn_finalize<<<1, 128, 0, stream>>>(g_sum, g_sumsq, gamma, beta, scale, shift,
                                     1.0f / (float)n_nodes);

  // 5. in-place BN apply + per-graph pooling (sorted graph ids -> register pooling)
  int waves = (n_nodes + 15) / 16;
  bn_apply_pool<<<(waves + 7) / 8, 256, 0, stream>>>(
      out_x, out_phis, scale, shift, gids, n_nodes);
}